// HierarchicalGraphNet_15796889715339
// MI455X (gfx1250) — compile-verified
//
#include <hip/hip_runtime.h>

typedef __attribute__((ext_vector_type(2))) float v2f;
typedef __attribute__((ext_vector_type(8))) float v8f;
typedef unsigned int u32x4 __attribute__((ext_vector_type(4)));
typedef int i32x4 __attribute__((ext_vector_type(4)));
typedef int i32x8 __attribute__((ext_vector_type(8)));

#define TPB 256
#define LDS_STRIDE 132   // 128 + 4 dword pad -> stride ≡ 4 (mod 64 banks): conflict-free b64 lane access

// ---------------------------------------------------------------------------
// Y[N x 128] = X[N x 128] @ W[128 x 128] (+ bias), fp32 via V_WMMA_F32_16X16X4_F32.
// One workgroup = 8 waves = 128 output rows.
//  - X tile (128x128 f32) staged into LDS by the Tensor Data Mover with the
//    pad feature giving a 132-float row stride (bank-conflict-free A frags).
//  - W staged column-major (stride 132) so each B fragment is one ds_load_b64.
// ---------------------------------------------------------------------------
__global__ __launch_bounds__(TPB)
void wmma_matmul_bias_kernel(const float* __restrict__ X,
                             const float* __restrict__ W,
                             const float* __restrict__ bias,
                             float* __restrict__ Y, int N)
{
    __shared__ __align__(16) float Wcm[128 * LDS_STRIDE]; // W^T, padded
    __shared__ __align__(16) float Xt [128 * LDS_STRIDE]; // X tile, padded by TDM

    const int tid = threadIdx.x;
    const int rowBase = blockIdx.x * 128;

    // ---- issue TDM load of this block's X tile (wave 0 only) ----
    if (tid < 32) {
        unsigned long long ga = (unsigned long long)(const void*)(X + (size_t)rowBase * 128);
        unsigned ldsoff = (unsigned)(unsigned long long)(void*)Xt;  // low 32 bits = LDS byte offset
        u32x4 g0;
        g0.x = 1u;                                   // count = 1 (valid user descriptor)
        g0.y = ldsoff;                               // lds_addr
        g0.z = (unsigned)ga;                         // global_addr[31:0]
        g0.w = (unsigned)((ga >> 32) & 0x1ffffffull) // global_addr[56:32]
             | 0x80000000u;                          // type = 2 ("image")
        int dim1 = N - rowBase;                      // rows left; TDM OOB reads return 0
        i32x8 g1;
        g1.s0 = (int)((2u << 16)    // data_size = 4B
                    | (1u << 20)    // pad_enable
                    | (6u << 22)    // pad_interval = 128 dwords
                    | (3u << 25));  // pad_amount   = 4 dwords  -> 132-float stride
        g1.s1 = (int)(128u << 16);                                  // tensor_dim0 = 128
        g1.s2 = (int)(((unsigned)dim1 & 0xffffu) << 16);            // dim0 hi=0 | tensor_dim1 lo
        g1.s3 = (int)(((unsigned)dim1 >> 16) | (128u << 16));       // tensor_dim1 hi | tile_dim0 = 128
        g1.s4 = 128;                                                // tile_dim1 = 128 (tile_dim2 = 0)
        g1.s5 = 128;                                                // tensor_dim0_stride = 128
        g1.s6 = 0;
        g1.s7 = 0;
        i32x4 z4 = { 0, 0, 0, 0 };
#if __clang_major__ >= 23
        i32x8 z8 = { 0, 0, 0, 0, 0, 0, 0, 0 };
        __builtin_amdgcn_tensor_load_to_lds(g0, g1, z4, z4, z8, 0);
#else
        __builtin_amdgcn_tensor_load_to_lds(g0, g1, z4, z4, 0);
#endif
    }

    // ---- cooperative transpose of W into padded column-major LDS (overlaps TDM) ----
    for (int idx = tid; idx < 128 * 128; idx += TPB) {
        int r = idx >> 7, c = idx & 127;
        Wcm[c * LDS_STRIDE + r] = W[idx];
    }
    if (tid < 32) __builtin_amdgcn_s_wait_tensorcnt(0);
    __syncthreads();

    const int wave = tid >> 5;
    const int lane = tid & 31;
    const int row0 = rowBase + wave * 16;
    if (row0 >= N) return;                 // wave-uniform exit

    const int lr = lane & 15;
    const int hk = (lane >> 4) << 1;       // lanes 0-15: K base +0, lanes 16-31: +2

    v8f zero = {};
    v8f acc[8];
#pragma unroll
    for (int c = 0; c < 8; ++c) acc[c] = zero;

    const int abase = (wave * 16 + lr) * LDS_STRIDE + hk;
    const int bbase = lr * LDS_STRIDE + hk;

#pragma unroll 4
    for (int ks = 0; ks < 32; ++ks) {
        float2 af = *(const float2*)&Xt[abase + 4 * ks];
        v2f a; a.x = af.x; a.y = af.y;
#pragma unroll
        for (int c = 0; c < 8; ++c) {
            float2 bf = *(const float2*)&Wcm[bbase + c * 16 * LDS_STRIDE + 4 * ks];
            v2f b; b.x = bf.x; b.y = bf.y;
            acc[c] = __builtin_amdgcn_wmma_f32_16x16x4_f32(
                false, a, false, b, (short)0, acc[c], false, false);
        }
    }

    const int mbase = row0 + ((lane >> 4) << 3);   // C/D: VGPR j -> row j (lo half) / j+8 (hi half)
#pragma unroll
    for (int c = 0; c < 8; ++c) {
        const int col = c * 16 + lr;
        const float bv = bias ? bias[col] : 0.0f;
#pragma unroll
        for (int j = 0; j < 8; ++j) {
            const int rr = mbase + j;
            if (rr < N) Y[(size_t)rr * 128 + col] = acc[c][j] + bv;
        }
    }
}

// ---------------------------------------------------------------------------
// Elementwise / edge kernels
// ---------------------------------------------------------------------------
__global__ void fill_f32(float* __restrict__ p, float v, long long n) {
    long long t = (long long)blockIdx.x * blockDim.x + threadIdx.x;
    if (t < n) p[t] = v;
}

__global__ void bias_bcast(float* __restrict__ out, const float* __restrict__ b, long long n) {
    long long t = (long long)blockIdx.x * blockDim.x + threadIdx.x;
    if (t < n) out[t] = b[t & 127];
}

__global__ void deg_accum(const int* __restrict__ col, const float* __restrict__ ew,
                          float* __restrict__ deg, int E, int useEw) {
    int e = blockIdx.x * blockDim.x + threadIdx.x;
    if (e < E) {
        float w = useEw ? ew[e] : 1.0f;
        atomicAdd(&deg[col[e]], w);
    }
}

__global__ void rsqrt_pos(float* __restrict__ p, int n) {
    int i = blockIdx.x * blockDim.x + threadIdx.x;
    if (i < n) {
        float d = p[i];
        p[i] = (d > 0.0f) ? rsqrtf(d) : 0.0f;
    }
}

__global__ void edge_norm(const int* __restrict__ row, const int* __restrict__ col,
                          const float* __restrict__ ew, const float* __restrict__ dinv,
                          float* __restrict__ nrm, int E, int useEw) {
    int e = blockIdx.x * blockDim.x + threadIdx.x;
    if (e < E) {
        float w = useEw ? ew[e] : 1.0f;
        nrm[e] = dinv[row[e]] * w * dinv[col[e]];
    }
}

// out[col[e], :] += h[row[e], :] * nrm[e]; one thread = one (edge, 4 features)
__global__ void aggregate_edges(const float* __restrict__ h, const int* __restrict__ row,
                                const int* __restrict__ col, const float* __restrict__ nrm,
                                float* __restrict__ out, long long work) {
    long long t = (long long)blockIdx.x * blockDim.x + threadIdx.x;
    if (t >= work) return;
    int e = (int)(t >> 5);
    int f = (int)(t & 31) << 2;
    const float* hp = h + (size_t)row[e] * 128 + f;
    __builtin_prefetch(hp, 0, 0);          // global_prefetch_b8
    float4 hv = *(const float4*)hp;
    float s = nrm[e];
    float* o = out + (size_t)col[e] * 128 + f;
    atomicAdd(o + 0, hv.x * s);
    atomicAdd(o + 1, hv.y * s);
    atomicAdd(o + 2, hv.z * s);
    atomicAdd(o + 3, hv.w * s);
}

__global__ void relu_ip(float* __restrict__ p, long long n) {
    long long t = (long long)blockIdx.x * blockDim.x + threadIdx.x;
    if (t < n) p[t] = fmaxf(p[t], 0.0f);
}

__global__ void seg_count(const int* __restrict__ cl, float* __restrict__ cnt, int N) {
    int i = blockIdx.x * blockDim.x + threadIdx.x;
    if (i < N) atomicAdd(&cnt[cl[i]], 1.0f);
}

__global__ void seg_sum(const float* __restrict__ x, const int* __restrict__ cl,
                        float* __restrict__ out, long long work) {
    long long t = (long long)blockIdx.x * blockDim.x + threadIdx.x;
    if (t >= work) return;
    int i = (int)(t >> 5);
    int f = (int)(t & 31) << 2;
    float4 v = *(const float4*)(x + (size_t)i * 128 + f);
    float* o = out + (size_t)cl[i] * 128 + f;
    atomicAdd(o + 0, v.x);
    atomicAdd(o + 1, v.y);
    atomicAdd(o + 2, v.z);
    atomicAdd(o + 3, v.w);
}

__global__ void seg_div(float* __restrict__ out, const float* __restrict__ cnt, long long n) {
    long long t = (long long)blockIdx.x * blockDim.x + threadIdx.x;
    if (t < n) out[t] = out[t] / fmaxf(cnt[t >> 7], 1.0f);
}

// out[i,:] = a[i,:] + b[cl[i],:]
__global__ void add_gather(const float* __restrict__ a, const float* __restrict__ b,
                           const int* __restrict__ cl, float* __restrict__ out, long long n) {
    long long t = (long long)blockIdx.x * blockDim.x + threadIdx.x;
    if (t < n) {
        int i = (int)(t >> 7);
        int f = (int)(t & 127);
        out[t] = a[t] + b[(size_t)cl[i] * 128 + f];
    }
}

// ---------------------------------------------------------------------------
// Host orchestration
// ---------------------------------------------------------------------------
static inline unsigned g1u(long long n) { return (unsigned)((n + TPB - 1) / TPB); }

extern "C" void kernel_launch(void* const* d_in, const int* in_sizes, int n_in,
                              void* d_out, int out_size, void* d_ws, size_t ws_size,
                              hipStream_t stream)
{
    const float* x     = (const float*)d_in[0];
    const float* W_pre = (const float*)d_in[1];
    const float* b_pre = (const float*)d_in[2];
    const float* W_u0  = (const float*)d_in[3];
    const float* b_u0  = (const float*)d_in[4];
    const float* W_u1  = (const float*)d_in[5];
    const float* b_u1  = (const float*)d_in[6];
    const float* W_u2  = (const float*)d_in[7];
    const float* b_u2  = (const float*)d_in[8];
    const float* W_d0  = (const float*)d_in[9];
    const float* b_d0  = (const float*)d_in[10];
    const float* W_d1  = (const float*)d_in[11];
    const float* b_d1  = (const float*)d_in[12];
    const int*   row0  = (const int*)d_in[13];
    const int*   col0  = (const int*)d_in[14];
    const float* ew0   = (const float*)d_in[15];
    const int*   row1  = (const int*)d_in[16];
    const int*   col1  = (const int*)d_in[17];
    const float* ew1   = (const float*)d_in[18];
    const int*   row2  = (const int*)d_in[19];
    const int*   col2  = (const int*)d_in[20];
    const float* ew2   = (const float*)d_in[21];
    const int*   cluster1 = (const int*)d_in[22];
    const int*   cluster2 = (const int*)d_in[23];

    const int N0 = in_sizes[0] / 128;
    const int C1 = in_sizes[23];      // len(cluster2) == C1
    const int C2 = 1000;              // fixed by the reference problem
    const int E0 = in_sizes[15], E1 = in_sizes[18], E2 = in_sizes[21];
    int Emax = E0 > E1 ? E0 : E1; if (E2 > Emax) Emax = E2;

    // --- workspace carve ---
    char* w = (char*)d_ws;
    auto carve = [&](size_t bytes) { void* p = (void*)w; w += (bytes + 255) & ~(size_t)255; return p; };
    float* A   = (float*)carve((size_t)N0 * 128 * 4);   // h / later y_n0
    float* Cb  = (float*)carve((size_t)N0 * 128 * 4);   // N0-level matmul scratch
    float* P   = (float*)carve((size_t)C1 * 128 * 4);   // x1 pre-mean / y
    float* Q   = (float*)carve((size_t)C1 * 128 * 4);   // x1 / y_c1
    float* R   = (float*)carve((size_t)C1 * 128 * 4);   // C1-level matmul scratch
    float* S0  = (float*)carve((size_t)C2 * 128 * 4);   // x2 pre
    float* S1  = (float*)carve((size_t)C2 * 128 * 4);   // x2
    float* T2  = (float*)carve((size_t)C2 * 128 * 4);   // C2-level matmul scratch
    float* deg = (float*)carve((size_t)N0 * 4);
    float* cnt = (float*)carve((size_t)C1 * 4);
    float* nrm = (float*)carve((size_t)Emax * 4);
    float* B   = (float*)d_out;                         // x0 lives in d_out until final pass

    auto mm = [&](const float* Xin, const float* Wm, const float* bv, float* Yout, int N) {
        int blocks = (N + 127) / 128;
        wmma_matmul_bias_kernel<<<blocks, TPB, 0, stream>>>(Xin, Wm, bv, Yout, N);
    };

    auto run_gcn = [&](const float* Xin, const float* Wm, const float* bvec,
                       const int* rw, const int* cl, const float* ew, int useEw,
                       int E, int N, float* scratch_mm, float* outp, int doRelu) {
        fill_f32<<<g1u(N), TPB, 0, stream>>>(deg, 0.0f, N);
        deg_accum<<<g1u(E), TPB, 0, stream>>>(cl, ew, deg, E, useEw);
        rsqrt_pos<<<g1u(N), TPB, 0, stream>>>(deg, N);
        edge_norm<<<g1u(E), TPB, 0, stream>>>(rw, cl, ew, deg, nrm, E, useEw);
        mm(Xin, Wm, nullptr, scratch_mm, N);
        bias_bcast<<<g1u((long long)N * 128), TPB, 0, stream>>>(outp, bvec, (long long)N * 128);
        aggregate_edges<<<g1u((long long)E * 32), TPB, 0, stream>>>(scratch_mm, rw, cl, nrm, outp, (long long)E * 32);
        if (doRelu)
            relu_ip<<<g1u((long long)N * 128), TPB, 0, stream>>>(outp, (long long)N * 128);
    };

    auto run_segmean = [&](const float* xin, const int* clu, int Nin, int Cout, float* outp) {
        fill_f32<<<g1u(Cout), TPB, 0, stream>>>(cnt, 0.0f, Cout);
        seg_count<<<g1u(Nin), TPB, 0, stream>>>(clu, cnt, Nin);
        fill_f32<<<g1u((long long)Cout * 128), TPB, 0, stream>>>(outp, 0.0f, (long long)Cout * 128);
        seg_sum<<<g1u((long long)Nin * 32), TPB, 0, stream>>>(xin, clu, outp, (long long)Nin * 32);
        seg_div<<<g1u((long long)Cout * 128), TPB, 0, stream>>>(outp, cnt, (long long)Cout * 128);
    };

    // 1) h = x @ W_pre + b_pre
    mm(x, W_pre, b_pre, A, N0);
    // 2) x0 = relu(GCN(h, W_u0, ew0))          -> B (= d_out)
    run_gcn(A, W_u0, b_u0, row0, col0, ew0, 1, E0, N0, Cb, B, 1);
    // 3) x1_pre = seg_mean(x0, cluster1)       -> P
    run_segmean(B, cluster1, N0, C1, P);
    // 4) x1 = relu(GCN(x1_pre, W_u1, ew1))     -> Q
    run_gcn(P, W_u1, b_u1, row1, col1, ew1, 1, E1, C1, R, Q, 1);
    // 5) x2 = relu(GCN(seg_mean(x1), W_u2))    -> S1
    run_segmean(Q, cluster2, C1, C2, S0);
    run_gcn(S0, W_u2, b_u2, row2, col2, ew2, 1, E2, C2, T2, S1, 1);
    // 6) y = x1 + x2[cluster2]                 -> P
    add_gather<<<g1u((long long)C1 * 128), TPB, 0, stream>>>(Q, S1, cluster2, P, (long long)C1 * 128);
    // 7) y = relu(GCN(y, W_d1, ones))          -> Q
    run_gcn(P, W_d1, b_d1, row1, col1, nullptr, 0, E1, C1, R, Q, 1);
    // 8) y = x0 + y[cluster1]                  -> A
    add_gather<<<g1u((long long)N0 * 128), TPB, 0, stream>>>(B, Q, cluster1, A, (long long)N0 * 128);
    // 9) out = GCN(y, W_d0, ones)              -> d_out
    run_gcn(A, W_d0, b_d0, row0, col0, nullptr, 0, E0, N0, Cb, B, 0);
}